// HierarchicalSoftmax_86930138071092
// MI455X (gfx1250) — compile-verified
//
#include <hip/hip_runtime.h>
#include <math.h>

#define EMBED     256
#define LVL_TILE  16
#define WARPS     8            // waves per block (wave32)
#define PSTRIDE   34           // LDS row stride in floats: conflict-free, 8B-aligned

typedef float v2f __attribute__((ext_vector_type(2)));
typedef float v8f __attribute__((ext_vector_type(8)));

// Stage 1: wave-per-sample gather + dot; WMMA(f32 16x16x4) reduces 16 levels x 32 lanes.
__global__ __launch_bounds__(WARPS * 32)
void hs_main(const float* __restrict__ emb,     // [N,256]
             const float* __restrict__ fc,      // [V-1,256]
             const int*   __restrict__ target,  // [N]
             const int*   __restrict__ pidx,    // [V,L]
             const float* __restrict__ pcode,   // [V,L]
             const float* __restrict__ pmask,   // [V,L]
             float2* __restrict__ ws,           // [gridDim.x] per-block partials
             int N, int L)
{
    __shared__ float  P[WARPS][LVL_TILE * PSTRIDE];  // partials [level][lane]
    __shared__ float2 CM[WARPS][LVL_TILE];           // (code, mask) per level
    __shared__ float2 wsum[WARPS];

    const int lane = threadIdx.x & 31;
    const int warp = threadIdx.x >> 5;
    const int n    = blockIdx.x * WARPS + warp;

    float accB = 0.0f;   // masked bce sum (live at lanes 0 and 16)
    float accC = 0.0f;   // mask count    (live at lanes 0 and 16)

    if (n < N) {
        const int t = target[n];
        __builtin_prefetch(&pidx[(size_t)t * L], 0, 0);   // global_prefetch_b8

        // Embedding row: 2 x b128 per lane; lane covers k = 4*lane..+3 and 128+4*lane..+3.
        const float4* e4 = (const float4*)(emb + (size_t)n * EMBED);
        const float4  e0 = e4[lane];
        const float4  e1 = e4[lane + 32];
        const float4* fc4 = (const float4*)fc;

        float*  Pw = P[warp];
        float2* cm = CM[warp];

        const int ntile = (L + LVL_TILE - 1) / LVL_TILE;
        for (int tb = 0; tb < ntile; ++tb) {
            const int l0 = tb * LVL_TILE;
            // Paths are front-packed: whole tile dead -> wave-uniform early exit.
            if (l0 >= L || pmask[(size_t)t * L + l0] == 0.0f) break;

            for (int lv = 0; lv < LVL_TILE; ++lv) {
                const int l = l0 + lv;
                int   idx  = 0;
                float code = 0.0f, m = 0.0f;
                if (l < L) {
                    const size_t o = (size_t)t * L + l;
                    idx  = pidx[o];
                    code = pcode[o];
                    m    = pmask[o];
                }
                // Gather one full fc row per wave: 2 x b128 per lane (1 KB, coalesced).
                const float4* w4 = fc4 + (size_t)idx * (EMBED / 4);
                const float4  w0 = w4[lane];
                const float4  w1 = w4[lane + 32];
                float p;
                p = e0.x * w0.x;
                p = fmaf(e0.y, w0.y, p);
                p = fmaf(e0.z, w0.z, p);
                p = fmaf(e0.w, w0.w, p);
                p = fmaf(e1.x, w1.x, p);
                p = fmaf(e1.y, w1.y, p);
                p = fmaf(e1.z, w1.z, p);
                p = fmaf(e1.w, w1.w, p);
                Pw[lv * PSTRIDE + lane] = p;                 // conflict-free ds_store_b32
                if (lane == 0) cm[lv] = make_float2(code, m);
            }

            // Same-wave LDS RAW: LDS is in-order per wave; wait + clobber stops reordering.
            asm volatile("s_wait_dscnt 0" ::: "memory");

            // Reduce-by-matmul: D[l][*] = sum_k P[l][k] * 1  (8 chained K=4 WMMAs).
            v8f acc = {};
            v2f bOnes; bOnes[0] = 1.0f; bOnes[1] = 1.0f;
            const int row  = lane & 15;
            const int koff = (lane >> 4) * 2;                // lanes 16-31 hold K=2,3
            #pragma unroll
            for (int kc = 0; kc < 8; ++kc) {
                v2f a = *(const v2f*)&Pw[row * PSTRIDE + kc * 4 + koff];
                acc = __builtin_amdgcn_wmma_f32_16x16x4_f32(
                        false, a, false, bOnes, (short)0, acc, false, false);
            }

            // D layout: lanes 0-15 -> pred[l0+0..7] in acc[0..7]; lanes 16-31 -> pred[l0+8..15].
            const int half = (lane >> 4) * 8;
            float sB = 0.0f, sC = 0.0f;
            #pragma unroll
            for (int v = 0; v < 8; ++v) {
                const float2 c = cm[v + half];               // broadcast ds_load_b64
                const float  p = acc[v];
                // bce_with_logits, stable: max(p,0) - p*c + log1p(exp(-|p|))
                const float bce = fmaxf(p, 0.0f) - p * c.x + log1pf(expf(-fabsf(p)));
                sB = fmaf(bce, c.y, sB);
                sC += c.y;
            }
            if ((lane & 15) == 0) { accB += sB; accC += sC; }  // lanes 0 and 16 only
            asm volatile("" ::: "memory");                     // keep reads before next tile's stores
        }

        // Fixed-tree wave reduction (only lanes 0 and 16 are nonzero).
        #pragma unroll
        for (int off = 16; off >= 1; off >>= 1) {
            accB += __shfl_xor(accB, off, 32);
            accC += __shfl_xor(accC, off, 32);
        }
    }

    if (lane == 0) wsum[warp] = make_float2(accB, accC);
    __syncthreads();
    if (threadIdx.x == 0) {
        float nB = 0.0f, nC = 0.0f;
        #pragma unroll
        for (int w = 0; w < WARPS; ++w) { nB += wsum[w].x; nC += wsum[w].y; }
        ws[blockIdx.x] = make_float2(nB, nC);
    }
}

// Stage 2: deterministic fixed-order reduction of per-block partials; out = sum/cnt.
__global__ __launch_bounds__(256)
void hs_reduce(const float2* __restrict__ ws, int nblk, float* __restrict__ out)
{
    __shared__ float sB[256];
    __shared__ float sC[256];
    const int tid = threadIdx.x;
    float b = 0.0f, c = 0.0f;
    for (int i = tid; i < nblk; i += 256) {
        const float2 v = ws[i];
        b += v.x; c += v.y;
    }
    sB[tid] = b; sC[tid] = c;
    __syncthreads();
    for (int s = 128; s > 0; s >>= 1) {
        if (tid < s) { sB[tid] += sB[tid + s]; sC[tid] += sC[tid + s]; }
        __syncthreads();
    }
    if (tid == 0) out[0] = sB[0] / sC[0];
}

extern "C" void kernel_launch(void* const* d_in, const int* in_sizes, int n_in,
                              void* d_out, int out_size, void* d_ws, size_t ws_size,
                              hipStream_t stream)
{
    const float* emb   = (const float*)d_in[0];   // [64,1,128,256]
    const float* fc    = (const float*)d_in[1];   // [V-1,256]
    const int*   tgt   = (const int*)d_in[2];     // [64,1,128]
    const int*   pidx  = (const int*)d_in[3];     // [V,L]
    const float* pcode = (const float*)d_in[4];   // [V,L]
    const float* pmask = (const float*)d_in[5];   // [V,L]

    const int N   = in_sizes[0] / EMBED;          // 8192 samples
    const int Vm1 = in_sizes[1] / EMBED;          // V-1
    const int V   = Vm1 + 1;
    const int L   = in_sizes[3] / V;              // padded path length

    const int nblk = (N + WARPS - 1) / WARPS;
    float2* ws = (float2*)d_ws;                   // nblk * 8 bytes

    hs_main<<<nblk, WARPS * 32, 0, stream>>>(emb, fc, tgt, pidx, pcode, pmask, ws, N, L);
    hs_reduce<<<1, 256, 0, stream>>>(ws, nblk, (float*)d_out);
}